// CorrectedNeoDTI_76459007804013
// MI455X (gfx1250) — compile-verified
//
#include <hip/hip_runtime.h>
#include <math.h>

#define NUM_DRUG 50000
#define NUM_CELL 20000
#define HDIM 128
#define NEDGE 600000
#define NBATCH 4096

typedef __attribute__((ext_vector_type(2))) float v2f;
typedef __attribute__((ext_vector_type(8))) float v8f;

// ---------------------------------------------------------------------------
// Per-destination edge counts (float, to match reference's f32 count divide)
// ---------------------------------------------------------------------------
__global__ void count_kernel(const int* __restrict__ dst, float* __restrict__ cnt, int ne) {
  int e = blockIdx.x * blockDim.x + threadIdx.x;
  if (e < ne) atomicAdd(&cnt[dst[e]], 1.0f);
}

__global__ void inv_kernel(float* __restrict__ cnt, int n) {
  int i = blockIdx.x * blockDim.x + threadIdx.x;
  if (i < n) cnt[i] = 1.0f / fmaxf(cnt[i], 1.0f);
}

// ---------------------------------------------------------------------------
// Scatter-add raw 128-d source embeddings into per-destination accumulators.
// 2 edges per 256-thread block; one f32 atomic per (edge, channel).
// ---------------------------------------------------------------------------
__global__ __launch_bounds__(256) void scatter_add_kernel(
    const float* __restrict__ h, const int* __restrict__ src,
    const int* __restrict__ dst, float* __restrict__ msg, int ne) {
  int e = blockIdx.x * 2 + (threadIdx.x >> 7);
  int k = threadIdx.x & 127;
  if (e < ne) {
    int s = src[e];
    int d = dst[e];
    atomicAdd(&msg[(size_t)d * HDIM + k], h[(size_t)s * HDIM + k]);
  }
}

// ---------------------------------------------------------------------------
// Fused node update: h_out = relu(layernorm(h_in + (msg_sum * inv_cnt) @ W^T))
// One block = 16 node rows. 8 waves, wave w owns output columns [16w,16w+16).
// GEMM via V_WMMA_F32_16X16X4_F32, K swept 0..128 in steps of 4.
// ---------------------------------------------------------------------------
__global__ __launch_bounds__(256) void update_kernel(
    const float* __restrict__ h_in, const float* __restrict__ msg,
    const float* __restrict__ invcnt, const float* __restrict__ Wmat,
    const float* __restrict__ gamma, const float* __restrict__ beta,
    float* __restrict__ h_out) {
  __shared__ float s_msg[16][HDIM];   // mean-scaled message tile (A)
  __shared__ float s_pre[16][HDIM];   // pre-layernorm rows
  __shared__ float s_sum[16][16];
  __shared__ float s_sq[16][16];

  const int tid = threadIdx.x;
  const int nodeBase = blockIdx.x * 16;

  // Stage A tile: 256 threads x 8 floats, scaled by 1/clamp(cnt,1)
  {
    const int r = tid >> 4;
    const int c0 = (tid & 15) * 8;
    const float inv = invcnt[nodeBase + r];
    const float4* p = (const float4*)&msg[(size_t)(nodeBase + r) * HDIM + c0];
    float4 a0 = p[0], a1 = p[1];
    s_msg[r][c0 + 0] = a0.x * inv; s_msg[r][c0 + 1] = a0.y * inv;
    s_msg[r][c0 + 2] = a0.z * inv; s_msg[r][c0 + 3] = a0.w * inv;
    s_msg[r][c0 + 4] = a1.x * inv; s_msg[r][c0 + 5] = a1.y * inv;
    s_msg[r][c0 + 6] = a1.z * inv; s_msg[r][c0 + 7] = a1.w * inv;
  }
  __syncthreads();

  const int wave = tid >> 5;          // 0..7 -> output column tile
  const int lane = tid & 31;
  const int m = lane & 15;            // A row within tile
  const int kb = (lane >> 4) * 2;     // K sub-offset {0,2}
  const int nGlob = wave * 16 + (lane & 15);  // output column (0..127)

  // A 16x4 f32 fragment: lane (m, hi) vgpr v -> A[m][4*step + 2*hi + v]
  // B 4x16 f32 fragment: lane (hi, n) vgpr v -> B[4*step + 2*hi + v][n] = W[n][k]
  v8f acc = {};
#pragma unroll
  for (int kk = 0; kk < HDIM; kk += 4) {
    v2f a = *(const v2f*)&s_msg[m][kk + kb];
    v2f b = *(const v2f*)&Wmat[(size_t)nGlob * HDIM + kk + kb];
    acc = __builtin_amdgcn_wmma_f32_16x16x4_f32(
        /*neg_a=*/false, a, /*neg_b=*/false, b,
        /*c_mod=*/(short)0, acc, /*reuse_a=*/false, /*reuse_b=*/false);
  }

  // C fragment: vgpr r, lanes 0-15 -> row r, lanes 16-31 -> row r+8; col = lane&15
  {
    const int mBase = (lane >> 4) * 8;
#pragma unroll
    for (int r = 0; r < 8; ++r) {
      const int mm = mBase + r;
      s_pre[mm][nGlob] = acc[r] + h_in[(size_t)(nodeBase + mm) * HDIM + nGlob];
    }
  }
  __syncthreads();

  // LayerNorm + ReLU: 16 threads per row, 8 elements each
  const int row = tid >> 4;
  const int sub = tid & 15;
  float s = 0.f, q = 0.f;
#pragma unroll
  for (int i = 0; i < 8; ++i) {
    float v = s_pre[row][sub * 8 + i];
    s += v;
    q += v * v;
  }
  s_sum[row][sub] = s;
  s_sq[row][sub] = q;
  __syncthreads();

  float ts = 0.f, tq = 0.f;
#pragma unroll
  for (int i = 0; i < 16; ++i) { ts += s_sum[row][i]; tq += s_sq[row][i]; }
  const float mean = ts * (1.0f / HDIM);
  const float var = tq * (1.0f / HDIM) - mean * mean;
  const float rstd = rsqrtf(var + 1e-5f);
#pragma unroll
  for (int i = 0; i < 8; ++i) {
    const int c = sub * 8 + i;
    float v = (s_pre[row][c] - mean) * rstd * gamma[c] + beta[c];
    h_out[(size_t)(nodeBase + row) * HDIM + c] = fmaxf(v, 0.0f);
  }
}

// ---------------------------------------------------------------------------
// Final head: logit = [h_drug[d]; h_cell[c]] . w + b -> sigmoid
// One wave32 per batch element, float4 loads, shuffle reduction.
// ---------------------------------------------------------------------------
__global__ __launch_bounds__(256) void final_kernel(
    const float* __restrict__ hd, const float* __restrict__ hc,
    const int* __restrict__ did, const int* __restrict__ cid,
    const float* __restrict__ wf, const float* __restrict__ wb,
    float* __restrict__ out, int nb) {
  int b = blockIdx.x * 8 + (threadIdx.x >> 5);
  int lane = threadIdx.x & 31;
  if (b >= nb) return;
  int d = did[b], c = cid[b];
  float4 x0 = *(const float4*)&hd[(size_t)d * HDIM + lane * 4];
  float4 w0 = *(const float4*)&wf[lane * 4];
  float4 x1 = *(const float4*)&hc[(size_t)c * HDIM + lane * 4];
  float4 w1 = *(const float4*)&wf[HDIM + lane * 4];
  float v = x0.x * w0.x + x0.y * w0.y + x0.z * w0.z + x0.w * w0.w +
            x1.x * w1.x + x1.y * w1.y + x1.z * w1.z + x1.w * w1.w;
#pragma unroll
  for (int off = 16; off; off >>= 1) v += __shfl_down(v, off, 32);
  if (lane == 0) out[b] = 1.0f / (1.0f + expf(-(v + wb[0])));
}

// ---------------------------------------------------------------------------
extern "C" void kernel_launch(void* const* d_in, const int* in_sizes, int n_in,
                              void* d_out, int out_size, void* d_ws, size_t ws_size,
                              hipStream_t stream) {
  const float* emb_drug = (const float*)d_in[0];
  const float* emb_cell = (const float*)d_in[1];
  const float* W_dt = (const float*)d_in[2];
  const float* W_td = (const float*)d_in[3];
  const float* ln_drug_g = (const float*)d_in[4];
  const float* ln_drug_b = (const float*)d_in[5];
  const float* ln_cell_g = (const float*)d_in[6];
  const float* ln_cell_b = (const float*)d_in[7];
  const float* W_final_w = (const float*)d_in[8];
  const float* W_final_b = (const float*)d_in[9];
  const int* edge_dt_src = (const int*)d_in[10];
  const int* edge_dt_dst = (const int*)d_in[11];
  const int* edge_td_src = (const int*)d_in[12];
  const int* edge_td_dst = (const int*)d_in[13];
  const int* drug_ids = (const int*)d_in[14];
  const int* cell_ids = (const int*)d_in[15];

  float* ws = (float*)d_ws;
  float* msg_drug = ws;                                  // 50000*128
  float* msg_cell = msg_drug + (size_t)NUM_DRUG * HDIM;  // 20000*128 (contiguous)
  float* inv_drug = msg_cell + (size_t)NUM_CELL * HDIM;  // 50000
  float* inv_cell = inv_drug + NUM_DRUG;                 // 20000 (contiguous)
  float* h_drug = inv_cell + NUM_CELL;                   // 50000*128
  float* h_cell = h_drug + (size_t)NUM_DRUG * HDIM;      // 20000*128

  // Edge counts (identical across layers): sum -> reciprocal of clamp(cnt,1)
  hipMemsetAsync(inv_drug, 0, (size_t)(NUM_DRUG + NUM_CELL) * sizeof(float), stream);
  count_kernel<<<(NEDGE + 255) / 256, 256, 0, stream>>>(edge_dt_dst, inv_cell, NEDGE);
  count_kernel<<<(NEDGE + 255) / 256, 256, 0, stream>>>(edge_td_dst, inv_drug, NEDGE);
  inv_kernel<<<(NUM_DRUG + NUM_CELL + 255) / 256, 256, 0, stream>>>(
      inv_drug, NUM_DRUG + NUM_CELL);

  for (int layer = 0; layer < 2; ++layer) {
    const float* hd_in = (layer == 0) ? emb_drug : h_drug;
    const float* hc_in = (layer == 0) ? emb_cell : h_cell;

    hipMemsetAsync(msg_drug, 0,
                   (size_t)(NUM_DRUG + NUM_CELL) * HDIM * sizeof(float), stream);
    // scatter_mean(X @ W^T) == scatter_mean(X) @ W^T : scatter raw embeddings
    scatter_add_kernel<<<(NEDGE + 1) / 2, 256, 0, stream>>>(
        hd_in, edge_dt_src, edge_dt_dst, msg_cell, NEDGE);
    scatter_add_kernel<<<(NEDGE + 1) / 2, 256, 0, stream>>>(
        hc_in, edge_td_src, edge_td_dst, msg_drug, NEDGE);

    update_kernel<<<NUM_CELL / 16, 256, 0, stream>>>(
        hc_in, msg_cell, inv_cell, W_dt, ln_cell_g, ln_cell_b, h_cell);
    update_kernel<<<NUM_DRUG / 16, 256, 0, stream>>>(
        hd_in, msg_drug, inv_drug, W_td, ln_drug_g, ln_drug_b, h_drug);
  }

  final_kernel<<<NBATCH / 8, 256, 0, stream>>>(
      h_drug, h_cell, drug_ids, cell_ids, W_final_w, W_final_b,
      (float*)d_out, NBATCH);
}